// skipgram_84439057039396
// MI455X (gfx1250) — compile-verified
//
#include <hip/hip_runtime.h>

// Skip-gram negative-sampling loss for MI455X (gfx1250, wave32).
// Memory-bound gather problem (~184 MB @ 23.3 TB/s ~ 8 us); dot products are
// computed 16-rows-at-a-time as f32 WMMA (V_WMMA_F32_16X16X4_F32) GEMMs whose
// diagonal holds the wanted scores. All data stays f32 (exact), each embedding
// row is streamed exactly once per use.
//
// Round-2 fixes vs round 1:
//  * chunk loop fully unrolled -> A tile stays in VGPRs (round 1 spilled the
//    64-VGPR A array to scratch and re-read it every inner iteration).
//  * branch-free diagonal extraction (v_cndmask chain) + a single log_sigmoid
//    per GEMM instead of 8 exec-masked copies.

typedef __attribute__((ext_vector_type(2))) float v2f;
typedef __attribute__((ext_vector_type(8))) float v8f;

#define DIM 128
#define NCHUNK (DIM / 4)   // K-chunks of 4 per 16x16x4 WMMA

__device__ __forceinline__ float log_sigmoid(float x) {
    // log(sigmoid(x)) = min(x,0) - log(1 + exp(-|x|))  (numerically stable)
    return fminf(x, 0.0f) - __logf(1.0f + __expf(-fabsf(x)));
}

__global__ __launch_bounds__(128) void skipgram_wmma_kernel(
    const float* __restrict__ u_emb,
    const float* __restrict__ v_emb,
    const int*   __restrict__ pos_u,
    const int*   __restrict__ pos_v,
    const int*   __restrict__ neg_v,
    float*       __restrict__ tile_out,
    int ntiles, int K)
{
    const int wave = threadIdx.x >> 5;
    const int lane = threadIdx.x & 31;
    const int tile = blockIdx.x * (blockDim.x >> 5) + wave;
    if (tile >= ntiles) return;

    const int half = lane >> 4;   // which K-pair of the 4-wide chunk this lane holds
    const int m    = lane & 15;   // A-row (M) / B-col (N) owned by this lane
    const int r0   = tile * 16;

    // Diagonal ownership: C[m,m] lives in component m on lane m (m<8),
    // or component m-8 on lane 16+m (m>=8)  => lanes 0-7 and 24-31.
    const bool diag_valid = (lane < 8) || (lane >= 24);
    const int  diag_comp  = (lane < 8) ? lane : (lane - 24);   // 0..7 (junk on invalid lanes)

    // ---- Load A tile (16x128 of u_emb) into WMMA-A register layout ----
    // A 16x4 f32 layout: lane L: M=L&15; VGPR0 = K=2*half, VGPR1 = K=2*half+1.
    const int urow = pos_u[r0 + m];
    const float* uptr = u_emb + (size_t)urow * DIM + 2 * half;
    v2f a[NCHUNK];
#pragma unroll
    for (int c = 0; c < NCHUNK; ++c)
        a[c] = *(const v2f*)(uptr + c * 4);   // 8B-aligned global_load_b64

    float acc = 0.0f;

    // ---- 1 positive + K negative GEMMs; scores are the C diagonal ----
    for (int s = 0; s <= K; ++s) {
        int vrow;
        if (s == 0) vrow = pos_v[r0 + m];
        else        vrow = neg_v[(size_t)(r0 + m) * K + (s - 1)];
        const float* vptr = v_emb + (size_t)vrow * DIM + 2 * half;

        v8f cacc = {};
#pragma unroll
        for (int c = 0; c < NCHUNK; ++c) {
            // B 4x16 f32 layout mirrors A: lane L: N=L&15; K = 2*half + {0,1}.
            v2f b = *(const v2f*)(vptr + c * 4);
            cacc = __builtin_amdgcn_wmma_f32_16x16x4_f32(
                false, a[c], false, b, (short)0, cacc, false, false);
        }

        // Branch-free diagonal pick: v_cndmask chain over the 8 C components.
        float d = cacc[0];
#pragma unroll
        for (int j = 1; j < 8; ++j)
            d = (diag_comp == j) ? cacc[j] : d;

        // pos pair: +score, negatives: -score; one log_sigmoid for all lanes,
        // masked accumulate on the 16 diagonal-owning lanes.
        const float sign = (s == 0) ? 1.0f : -1.0f;
        const float contrib = log_sigmoid(sign * d);
        acc += diag_valid ? contrib : 0.0f;
    }

    // ---- wave32 reduction of the 16 per-row contributions ----
#pragma unroll
    for (int off = 16; off > 0; off >>= 1)
        acc += __shfl_xor(acc, off, 32);
    if (lane == 0) tile_out[tile] = acc;
}

__global__ __launch_bounds__(256) void skipgram_reduce_kernel(
    const float* __restrict__ tile_out, float* __restrict__ out,
    int ntiles, float invB)
{
    __shared__ float sdata[256];
    float s = 0.0f;
    for (int i = threadIdx.x; i < ntiles; i += 256) s += tile_out[i];
    sdata[threadIdx.x] = s;
    __syncthreads();
#pragma unroll
    for (int stride = 128; stride > 0; stride >>= 1) {
        if (threadIdx.x < (unsigned)stride)
            sdata[threadIdx.x] += sdata[threadIdx.x + stride];
        __syncthreads();
    }
    if (threadIdx.x == 0) out[0] = -sdata[0] * invB;   // loss = -sum/B
}

extern "C" void kernel_launch(void* const* d_in, const int* in_sizes, int n_in,
                              void* d_out, int out_size, void* d_ws, size_t ws_size,
                              hipStream_t stream) {
    const float* u_emb = (const float*)d_in[0];
    const float* v_emb = (const float*)d_in[1];
    const int*   pos_u = (const int*)d_in[2];
    const int*   pos_v = (const int*)d_in[3];
    const int*   neg_v = (const int*)d_in[4];

    const int B = in_sizes[2];             // 16384
    const int K = in_sizes[4] / B;         // 20
    const int ntiles = B / 16;             // 1024 (B is a multiple of 16)

    float* tile_out = (float*)d_ws;        // ntiles floats of scratch

    const int waves_per_block = 4;         // 128 threads -> 256 blocks, good WGP spread
    const int blocks = (ntiles + waves_per_block - 1) / waves_per_block;

    skipgram_wmma_kernel<<<blocks, waves_per_block * 32, 0, stream>>>(
        u_emb, v_emb, pos_u, pos_v, neg_v, tile_out, ntiles, K);
    skipgram_reduce_kernel<<<1, 256, 0, stream>>>(
        tile_out, (float*)d_out, ntiles, 1.0f / (float)B);
}